// MultiHeadAttention_88278757802063
// MI455X (gfx1250) — compile-verified
//
#include <hip/hip_runtime.h>

// ---------------------------------------------------------------------------
// MultiHeadAttention for MI455X (gfx1250), fp32 via V_WMMA_F32_16X16X4_F32.
// B=2, S=2048, D=1024, H=16, HD=64.
// d_out = [out (B,S,D)] ++ [attn (B,H,S,S)]
//
// GEMMs use double-buffered LDS fed by GLOBAL_LOAD_ASYNC_TO_LDS_B128
// (ASYNCcnt-tracked DMA, no VGPR round-trip) overlapping the next K-chunk's
// loads with the current chunk's WMMA work.
// ---------------------------------------------------------------------------

typedef __attribute__((ext_vector_type(2))) float v2f;
typedef __attribute__((ext_vector_type(8))) float v8f;
typedef __attribute__((__vector_size__(16))) int v4i_;

#define B_ 2
#define S_ 2048
#define D_ 1024
#define H_ 16
#define HD_ 64
#define NROW_ (B_ * S_)   // 4096

static __device__ __forceinline__ v8f wmma_f32(v2f a, v2f b, v8f c) {
  // D = A(16x4, f32) * B(4x16, f32) + C(16x16, f32)
  return __builtin_amdgcn_wmma_f32_16x16x4_f32(
      /*neg_a=*/false, a, /*neg_b=*/false, b,
      /*c_mod=*/(short)0, c, /*reuse_a=*/false, /*reuse_b=*/false);
}

// Async DMA: 16 bytes global -> LDS per lane, tracked by ASYNCcnt.
static __device__ __forceinline__ void async_copy_b128(const float* g, float* l) {
  __builtin_amdgcn_global_load_async_to_lds_b128(
      (__attribute__((address_space(1))) v4i_*)g,
      (__attribute__((address_space(3))) v4i_*)l, 0, 0);
}

static __device__ __forceinline__ void wait_asynccnt0() {
  asm volatile("s_wait_asynccnt 0x0" ::: "memory");
}

// ---------------------------------------------------------------------------
// GEMM: Y[M,N] = X[M,K] @ W[N,K]^T + bias[N]
// Block: 256 threads (8 waves). Block tile 128x256, wave tile 64x64.
// K chunks of 16 staged through double-buffered LDS via async DMA.
// ---------------------------------------------------------------------------
#define BM 128
#define BN 256
#define BK 16
#define LDP 20   // padded LDS row stride (floats); 20*4B keeps 16B alignment

__global__ __launch_bounds__(256) void gemm_xwT_bias(
    const float* __restrict__ X, const float* __restrict__ W,
    const float* __restrict__ bias, float* __restrict__ Y,
    int M, int N, int K) {
  __shared__ float sX[2][BM * LDP];  // 2 x 10 KB
  __shared__ float sW[2][BN * LDP];  // 2 x 20 KB

  const int tid  = threadIdx.x;
  const int wave = tid >> 5;
  const int lane = tid & 31;
  const int lm   = lane & 15;   // N / M row select within fragments
  const int lh   = lane >> 4;   // half-wave select (K striping)
  const int bm   = blockIdx.x * BM;
  const int bn   = blockIdx.y * BN;
  const int wm   = (wave >> 2) * 64;  // 0 or 64
  const int wn   = (wave & 3) * 64;   // 0..192

  // Per-thread staging coordinates (16B per async issue).
  const int r2[2] = {(tid + 0) >> 2, (tid + 256) >> 2};
  const int c2[2] = {((tid + 0) & 3) << 2, ((tid + 256) & 3) << 2};

  v8f acc[4][4];
#pragma unroll
  for (int i = 0; i < 4; ++i)
#pragma unroll
    for (int j = 0; j < 4; ++j) acc[i][j] = (v8f)0.0f;

  auto stage = [&](int buf, int kc) {
    // X tile: 128x16 floats = 512 x b128, 2 issues/thread.
#pragma unroll
    for (int i = 0; i < 2; ++i)
      async_copy_b128(&X[(size_t)(bm + r2[i]) * K + kc + c2[i]],
                      &sX[buf][r2[i] * LDP + c2[i]]);
    // W tile: 256x16 floats = 1024 x b128, 4 issues/thread.
#pragma unroll
    for (int i = 0; i < 4; ++i) {
      int linear = tid + i * 256;
      int r = linear >> 2, c4 = (linear & 3) << 2;
      async_copy_b128(&W[(size_t)(bn + r) * K + kc + c4],
                      &sW[buf][r * LDP + c4]);
    }
  };

  stage(0, 0);
  const int niter = K / BK;
  for (int it = 0; it < niter; ++it) {
    const int cur = it & 1;
    const int kc  = it * BK;
    wait_asynccnt0();    // this wave's async loads for buf[cur] have landed
    __syncthreads();     // all waves: data visible AND prior reads of buf[cur^1] done
    if (it + 1 < niter) stage(cur ^ 1, kc + BK);  // overlap DMA with compute

#pragma unroll
    for (int s = 0; s < BK; s += 4) {
      const int kk = s + 2 * lh;  // this lane-half's K pair within the chunk
      v2f a[4], b[4];
#pragma unroll
      for (int i = 0; i < 4; ++i)
        a[i] = *(const v2f*)&sX[cur][(wm + i * 16 + lm) * LDP + kk];
#pragma unroll
      for (int j = 0; j < 4; ++j)
        b[j] = *(const v2f*)&sW[cur][(wn + j * 16 + lm) * LDP + kk];
#pragma unroll
      for (int i = 0; i < 4; ++i)
#pragma unroll
        for (int j = 0; j < 4; ++j)
          acc[i][j] = wmma_f32(a[i], b[j], acc[i][j]);
    }
  }

  // Epilogue: C/D layout -> m = r + 8*lh, n = lm within each 16x16 tile.
#pragma unroll
  for (int j = 0; j < 4; ++j) {
    const int n  = bn + wn + j * 16 + lm;
    const float bv = bias[n];
#pragma unroll
    for (int i = 0; i < 4; ++i) {
#pragma unroll
      for (int r = 0; r < 8; ++r) {
        const int m = bm + wm + i * 16 + r + 8 * lh;
        Y[(size_t)m * N + n] = acc[i][j][r] + bv;
      }
    }
  }
}

// ---------------------------------------------------------------------------
// Fused attention: one workgroup per (b, h, 16-query-row block).
// Phase 1: scores[16][2048] = scale * Q Kt  (WMMA, K=HD=64)  -> LDS
// Phase 2: exact softmax per row (wave shfl reductions), write attn (NT stores)
// Phase 3: ctx[16][64] = P @ V (WMMA, K=2048 split over 8 waves) + LDS reduce
// ---------------------------------------------------------------------------
#define SW 2052  // padded score row stride (floats) -> conflict-free phases

__global__ __launch_bounds__(256) void attn_fused(
    const float* __restrict__ Q, const float* __restrict__ Kmat,
    const float* __restrict__ V, float* __restrict__ attn,
    float* __restrict__ ctx) {
  __shared__ float sS[16 * SW];  // 128.25 KB scores / P; reused for partials

  const int tid  = threadIdx.x;
  const int wave = tid >> 5;
  const int lane = tid & 31;
  const int lm   = lane & 15;
  const int lh   = lane >> 4;

  const int q0 = blockIdx.x * 16;  // query row block
  const int h  = blockIdx.y;
  const int b  = blockIdx.z;
  const size_t qkv_row0 = (size_t)(b * S_) * D_;  // base of this batch
  const int hc = h * HD_;                          // head column offset

  // ---- Phase 1: scores ----------------------------------------------------
  // A fragments (Q rows q0..q0+15, all of HD=64) held in registers, reused
  // across every key tile this wave processes.
  v2f af[16];
  {
    const float* qrow = Q + qkv_row0 + (size_t)(q0 + lm) * D_ + hc;
#pragma unroll
    for (int s = 0; s < 16; ++s) af[s] = *(const v2f*)&qrow[s * 4 + 2 * lh];
  }

  const float scale = 0.125f;  // 1/sqrt(64)
  for (int t = 0; t < 16; ++t) {
    const int c0 = wave * 256 + t * 16;  // key column tile
    const float* krow = Kmat + qkv_row0 + (size_t)(c0 + lm) * D_ + hc;
    v8f acc = (v8f)0.0f;
#pragma unroll
    for (int s = 0; s < 16; ++s) {
      v2f bf = *(const v2f*)&krow[s * 4 + 2 * lh];
      acc = wmma_f32(af[s], bf, acc);
    }
#pragma unroll
    for (int r = 0; r < 8; ++r)
      sS[(r + 8 * lh) * SW + c0 + lm] = acc[r] * scale;
  }
  __syncthreads();

  // ---- Phase 2: softmax (2 rows per wave; deterministic reductions) -------
  {
    for (int rr = 0; rr < 2; ++rr) {
      const int row = wave * 2 + rr;
      float* srow = &sS[row * SW];
      float m = -3.402823466e+38f;
      for (int c = lane; c < S_; c += 32) m = fmaxf(m, srow[c]);
#pragma unroll
      for (int off = 16; off > 0; off >>= 1)
        m = fmaxf(m, __shfl_xor(m, off, 32));
      float sum = 0.0f;
      for (int c = lane; c < S_; c += 32) {
        float e = __expf(srow[c] - m);
        srow[c] = e;
        sum += e;
      }
#pragma unroll
      for (int off = 16; off > 0; off >>= 1) sum += __shfl_xor(sum, off, 32);
      const float inv = 1.0f / sum;
      float* arow =
          attn + (((size_t)(b * H_ + h) * S_) + (q0 + row)) * (size_t)S_;
      for (int c = lane; c < S_; c += 32) {
        float p = srow[c] * inv;
        srow[c] = p;
        __builtin_nontemporal_store(p, &arow[c]);  // 537 MB stream: bypass
      }
    }
  }
  __syncthreads();

  // ---- Phase 3: ctx = P @ V, K split across waves --------------------------
  const int koff = wave * 256;
  v8f cacc[4];
#pragma unroll
  for (int j = 0; j < 4; ++j) cacc[j] = (v8f)0.0f;

  for (int s = 0; s < 64; ++s) {
    const int kk = koff + s * 4 + 2 * lh;  // this lane-half's K pair
    v2f a = *(const v2f*)&sS[lm * SW + kk];
    const float* vr0 = V + qkv_row0 + (size_t)kk * D_ + hc;
    const float* vr1 = vr0 + D_;
#pragma unroll
    for (int j = 0; j < 4; ++j) {
      v2f bf;
      bf.x = vr0[j * 16 + lm];
      bf.y = vr1[j * 16 + lm];
      cacc[j] = wmma_f32(a, bf, cacc[j]);
    }
  }
  __syncthreads();  // all P reads complete; safe to reuse sS for partials

  // partials: [wave][16][64]
#pragma unroll
  for (int j = 0; j < 4; ++j)
#pragma unroll
    for (int r = 0; r < 8; ++r)
      sS[(wave * 16 + r + 8 * lh) * 64 + j * 16 + lm] = cacc[j][r];
  __syncthreads();

  // Fixed-order cross-wave reduction, then write ctx (deterministic).
  for (int e = tid; e < 16 * 64; e += 256) {
    const int m = e >> 6, n = e & 63;
    float sum = 0.0f;
#pragma unroll
    for (int w = 0; w < 8; ++w) sum += sS[(w * 16 + m) * 64 + n];
    ctx[qkv_row0 + (size_t)(q0 + m) * D_ + hc + n] = sum;
  }
}

// ---------------------------------------------------------------------------
extern "C" void kernel_launch(void* const* d_in, const int* in_sizes, int n_in,
                              void* d_out, int out_size, void* d_ws,
                              size_t ws_size, hipStream_t stream) {
  (void)in_sizes; (void)n_in; (void)out_size; (void)ws_size;
  const float* x  = (const float*)d_in[0];
  const float* Wq = (const float*)d_in[1];
  const float* bq = (const float*)d_in[2];
  const float* Wk = (const float*)d_in[3];
  const float* bk = (const float*)d_in[4];
  const float* Wv = (const float*)d_in[5];
  const float* bv = (const float*)d_in[6];
  const float* Wo = (const float*)d_in[7];
  const float* bo = (const float*)d_in[8];

  float* out  = (float*)d_out;                       // [B,S,D]
  float* attn = out + (size_t)B_ * S_ * D_;          // [B,H,S,S]

  float* Qb = (float*)d_ws;                          // 16 MB each
  float* Kb = Qb + (size_t)NROW_ * D_;
  float* Vb = Kb + (size_t)NROW_ * D_;
  float* Cb = Vb + (size_t)NROW_ * D_;               // needs 64 MB workspace

  const dim3 gg(NROW_ / BM, D_ / BN), bb(256);
  gemm_xwT_bias<<<gg, bb, 0, stream>>>(x, Wq, bq, Qb, NROW_, D_, D_);
  gemm_xwT_bias<<<gg, bb, 0, stream>>>(x, Wk, bk, Kb, NROW_, D_, D_);
  gemm_xwT_bias<<<gg, bb, 0, stream>>>(x, Wv, bv, Vb, NROW_, D_, D_);
  attn_fused<<<dim3(S_ / 16, H_, B_), 256, 0, stream>>>(Qb, Kb, Vb, attn, Cb);
  gemm_xwT_bias<<<gg, bb, 0, stream>>>(Cb, Wo, bo, out, NROW_, D_, D_);
}